// CrossAttentionBlock_46651934769202
// MI455X (gfx1250) — compile-verified
//
#include <hip/hip_runtime.h>

typedef __attribute__((ext_vector_type(2))) float v2f;
typedef __attribute__((ext_vector_type(4))) float v4f;
typedef __attribute__((ext_vector_type(8))) float v8f;

#define HIDDEN 128

// C[m,n] = sum_k A[m,k] * W[n,k] + bias[n]
// A: M x K row-major, W: N x K row-major (so B = W^T), C: M x N row-major.
// One wave computes one 16x16 tile via V_WMMA_F32_16X16X4_F32.
// VGPR layouts per CDNA5 ISA 7.12.2:
//   A 16x4 f32 : lanes 0-15 -> M=lane, {v0=K0, v1=K1}; lanes 16-31 -> {v0=K2, v1=K3}
//   B 4x16 f32 : lanes 0-15 -> N=lane, {v0=K0, v1=K1}; lanes 16-31 -> {v0=K2, v1=K3}
//   C/D 16x16  : vgpr r, lanes 0-15 -> (M=r, N=lane); lanes 16-31 -> (M=r+8, N=lane-16)
__global__ void gemm_nt_f32_wmma(const float* __restrict__ A,
                                 const float* __restrict__ W,
                                 const float* __restrict__ bias,
                                 float* __restrict__ C,
                                 int M, int N, int K) {
  const int lane = threadIdx.x & 31;
  const int wave = threadIdx.x >> 5;
  const int tile = blockIdx.x * (blockDim.x >> 5) + wave;
  const int ntiles_n = N >> 4;
  const int mt = tile / ntiles_n;
  const int nt = tile - mt * ntiles_n;
  if (mt * 16 >= M) return;                       // wave-uniform

  const int half = lane >> 4;                     // 0: K pair {0,1}, 1: K pair {2,3}
  const int l    = lane & 15;
  const float* __restrict__ arow = A + (size_t)(mt * 16 + l) * K + 2 * half;
  const float* __restrict__ wrow = W + (size_t)(nt * 16 + l) * K + 2 * half;

  v8f acc = {};
  for (int k0 = 0; k0 < K; k0 += 4) {
    v2f a, b;
    a.x = arow[k0];  a.y = arow[k0 + 1];          // one global_load_b64
    b.x = wrow[k0];  b.y = wrow[k0 + 1];          // one global_load_b64
    acc = __builtin_amdgcn_wmma_f32_16x16x4_f32(
        /*neg_a=*/false, a, /*neg_b=*/false, b,
        /*c_mod=*/(short)0, acc, /*reuse_a=*/false, /*reuse_b=*/false);
  }

  const float bn = bias[nt * 16 + l];
#pragma unroll
  for (int r = 0; r < 8; ++r) {
    C[(size_t)(mt * 16 + r + 8 * half) * N + nt * 16 + l] = acc[r] + bn;
  }
}

// Row-wise LayerNorm over rows of length 128. One 128-thread block per row.
__global__ void layernorm_rows(const float* __restrict__ X,
                               const float* __restrict__ gamma,
                               const float* __restrict__ beta,
                               float* __restrict__ Y) {
  const int row = blockIdx.x;
  const int t   = threadIdx.x;                    // 0..127
  __shared__ float red[HIDDEN];

  const float x = X[row * HIDDEN + t];
  red[t] = x;
  __syncthreads();
#pragma unroll
  for (int s = 64; s > 0; s >>= 1) {
    if (t < s) red[t] += red[t + s];
    __syncthreads();
  }
  const float mu = red[0] * (1.0f / HIDDEN);
  __syncthreads();

  const float d = x - mu;
  red[t] = d * d;
  __syncthreads();
#pragma unroll
  for (int s = 64; s > 0; s >>= 1) {
    if (t < s) red[t] += red[t + s];
    __syncthreads();
  }
  const float var = red[0] * (1.0f / HIDDEN);
  const float inv = rsqrtf(var + 1e-5f);
  Y[row * HIDDEN + t] = d * inv * gamma[t] + beta[t];
}

// out[n,:] = table[idx[n],:]  (128 floats per row).
// One wave per node: lane l moves float4 at column 4*l -> 512B coalesced NT store.
__global__ void bcast_rows(const v4f* __restrict__ table,
                           const int* __restrict__ idx,
                           v4f* __restrict__ out,
                           int n_nodes) {
  const int gwave = (int)((blockIdx.x * blockDim.x + threadIdx.x) >> 5);
  const int lane  = threadIdx.x & 31;
  if (gwave >= n_nodes) return;
  const int row = idx[gwave];                     // L2/L0-resident table
  const v4f val = table[(size_t)row * 32 + lane];
  __builtin_nontemporal_store(val, &out[(size_t)gwave * 32 + lane]);
}

extern "C" void kernel_launch(void* const* d_in, const int* in_sizes, int n_in,
                              void* d_out, int out_size, void* d_ws, size_t ws_size,
                              hipStream_t stream) {
  // setup_inputs order:
  // 0 masked_x (unused), 1 graph_vec, 2 batch_indices, 3 Wq(u), 4 bq(u),
  // 5 Wk(u), 6 bk(u), 7 Wv, 8 bv, 9 Win, 10 bin, 11 Wout, 12 bout, 13 gamma, 14 beta
  const float* graph_vec = (const float*)d_in[1];   // 128 x 256
  const int*   bidx      = (const int*)d_in[2];     // 400000
  const float* Wv        = (const float*)d_in[7];   // 128 x 256
  const float* bv        = (const float*)d_in[8];   // 128
  const float* Win       = (const float*)d_in[9];   // 384 x 128
  const float* bin_      = (const float*)d_in[10];  // 384
  const float* Wout      = (const float*)d_in[11];  // 128 x 128
  const float* bout      = (const float*)d_in[12];  // 128
  const float* gamma     = (const float*)d_in[13];  // 128
  const float* beta      = (const float*)d_in[14];  // 128

  const int n_nodes = in_sizes[2];                  // 400000
  const int GRAPH_D = in_sizes[1] / HIDDEN;         // 256 (BATCH=128 rows)

  const float* Wiv = Win + 2 * HIDDEN * HIDDEN;     // Win[256:384]
  const float* biv = bin_ + 2 * HIDDEN;

  float* ws    = (float*)d_ws;
  float* t_v   = ws;                                // 128*128
  float* t_v2  = ws + 1 * HIDDEN * HIDDEN;
  float* t_att = ws + 2 * HIDDEN * HIDDEN;
  float* t_ln  = ws + 3 * HIDDEN * HIDDEN;

  // 64 tiles of 16x16 -> 16 blocks x 4 waves (128 threads) per GEMM.
  dim3 gblk(128), ggrd(16);
  gemm_nt_f32_wmma<<<ggrd, gblk, 0, stream>>>(graph_vec, Wv,  bv,  t_v,   HIDDEN, HIDDEN, GRAPH_D);
  gemm_nt_f32_wmma<<<ggrd, gblk, 0, stream>>>(t_v,       Wiv, biv, t_v2,  HIDDEN, HIDDEN, HIDDEN);
  gemm_nt_f32_wmma<<<ggrd, gblk, 0, stream>>>(t_v2,      Wout, bout, t_att, HIDDEN, HIDDEN, HIDDEN);

  layernorm_rows<<<dim3(HIDDEN), dim3(HIDDEN), 0, stream>>>(t_att, gamma, beta, t_ln);

  // One wave (32 lanes) per node, 256 threads = 8 waves per block.
  const int waves_per_block = 256 / 32;
  const int nblocks = (n_nodes + waves_per_block - 1) / waves_per_block;
  bcast_rows<<<dim3(nblocks), dim3(256), 0, stream>>>((const v4f*)t_ln, bidx,
                                                      (v4f*)d_out, n_nodes);
}